// VAE_head_1949915152483
// MI455X (gfx1250) — compile-verified
//
#include <hip/hip_runtime.h>
#include <math.h>

typedef __attribute__((ext_vector_type(2))) float v2f;
typedef __attribute__((ext_vector_type(4))) float v4f;
typedef __attribute__((ext_vector_type(8))) float v8f;

#define B_ 8192
#define D_ 512
#define H_ 1024
#define S_ 4096
#define Y_ 256

#define BM 128
#define BN 64
#define BK 16

// ---------------------------------------------------------------------------
// Tiled fp32 WMMA GEMM: C[M,N] = A[M,K] @ B (+ epilogue)
//   BT=false: B is [K,N] row-major (weights)
//   BT=true : B is [N,K] row-major (prototypes; effective B = B^T)
// EPI: 0 = +bias, 1 = relu(+bias), 2 = 2*v - rown[row] - coln[col], 3 = raw dot
// Block: 256 threads (8 waves). Each wave: 16 rows x 64 cols (4 accumulators).
// Staging: one b128 global load -> one b128 LDS store per tile slot, software
// pipelined: next tile's global loads issue while WMMA consumes current LDS.
// A tile (and BT B tile) use row stride 20 floats: 16B-aligned b128 stores AND
// conflict-free column reads (r*20 mod 64 distinct for r=0..15).
// ---------------------------------------------------------------------------
template <int EPI, bool BT>
__global__ __launch_bounds__(256) void gemm_wmma_f32(
    const float* __restrict__ A, const float* __restrict__ Bm,
    const float* __restrict__ bias, float* __restrict__ C,
    int M, int N, int K,
    const float* __restrict__ rown, const float* __restrict__ coln)
{
    __shared__ __align__(16) float As[BM][BK + 4];         // 128 x 20
    constexpr int BR = BT ? BN : BK;
    constexpr int BCc = BT ? (BK + 4) : (BN + 4);
    __shared__ __align__(16) float Bs[BR][BCc];            // 64x20 or 16x68

    const int tid  = threadIdx.x;
    const int lane = tid & 31;
    const int wv   = tid >> 5;
    const int hi   = lane >> 4;      // 0 for lanes 0-15, 1 for lanes 16-31
    const int l16  = lane & 15;
    const int m0   = blockIdx.x * BM;
    const int n0   = blockIdx.y * BN;

    // ---- staging assignments (computed once) ----
    // A: 128x16 = 512 float4 slots, 2 per thread
    const int ar0 = tid >> 2,          ac0 = (tid & 3) * 4;
    const int ar1 = (tid + 256) >> 2,  ac1 = ((tid + 256) & 3) * 4;
    const float* aptr0 = A + (size_t)(m0 + ar0) * K + ac0;
    const float* aptr1 = A + (size_t)(m0 + ar1) * K + ac1;
    // B: 16x64 = 256 float4 slots, 1 per thread
    int br, bc;
    const float* bptr;
    size_t bstride;
    if (BT) { br = tid >> 2;  bc = (tid & 3) * 4;
              bptr = Bm + (size_t)(n0 + br) * K + bc;      bstride = BK; }
    else    { br = tid >> 4;  bc = (tid & 15) * 4;
              bptr = Bm + (size_t)br * N + n0 + bc;        bstride = (size_t)BK * N; }

    v8f acc[4];
    for (int t = 0; t < 4; ++t)
        for (int e = 0; e < 8; ++e) acc[t][e] = 0.0f;

    // pipeline prologue: first tile loads
    v4f a0v = *(const v4f*)aptr0;  aptr0 += BK;
    v4f a1v = *(const v4f*)aptr1;  aptr1 += BK;
    v4f bvv = *(const v4f*)bptr;   bptr  += bstride;

    for (int k0 = 0; k0 < K; k0 += BK) {
        *(v4f*)&As[ar0][ac0] = a0v;
        *(v4f*)&As[ar1][ac1] = a1v;
        *(v4f*)&Bs[br][bc]   = bvv;
        __syncthreads();

        // prefetch next tile (issues during WMMA compute below)
        if (k0 + BK < K) {
            a0v = *(const v4f*)aptr0;  aptr0 += BK;
            a1v = *(const v4f*)aptr1;  aptr1 += BK;
            bvv = *(const v4f*)bptr;   bptr  += bstride;
        }

        // 16x16x4 fp32 WMMA, 4 k-steps per 16-chunk
        for (int kk = 0; kk < BK; kk += 4) {
            const int ak = kk + 2 * hi;
            v2f a = *(const v2f*)&As[wv * 16 + l16][ak];
            for (int t = 0; t < 4; ++t) {
                v2f b;
                if (BT) {
                    b = *(const v2f*)&Bs[t * 16 + l16][ak];
                } else {
                    b.x = Bs[ak][t * 16 + l16];
                    b.y = Bs[ak + 1][t * 16 + l16];
                }
                acc[t] = __builtin_amdgcn_wmma_f32_16x16x4_f32(
                    false, a, false, b, (short)0, acc[t], false, false);
            }
        }
        __syncthreads();
    }

    // epilogue: C/D layout -> VGPR r holds row (r + 8*hi), col = l16 (+16*t)
    const int row_base = m0 + wv * 16 + hi * 8;
    for (int t = 0; t < 4; ++t) {
        int col = n0 + t * 16 + l16;
        float bv2 = 0.0f, cn = 0.0f;
        if (EPI == 0 || EPI == 1) bv2 = bias[col];
        if (EPI == 2)             cn = coln[col];
        for (int r = 0; r < 8; ++r) {
            int row = row_base + r;
            float v = acc[t][r];
            float o;
            if (EPI == 0)      o = v + bv2;
            else if (EPI == 1) { o = v + bv2; o = o > 0.0f ? o : 0.0f; }
            else if (EPI == 2) o = 2.0f * v - rown[row] - cn;
            else               o = v;
            C[(size_t)row * N + col] = o;
        }
    }
}

// ---------------------------------------------------------------------------
// Z = mu + softplus(spre) * eps   (float4 per lane)
// ---------------------------------------------------------------------------
__global__ __launch_bounds__(256) void z_kernel(
    const float* __restrict__ mu, const float* __restrict__ spre,
    const float* __restrict__ eps, float* __restrict__ Z, int n4)
{
    int i = blockIdx.x * blockDim.x + threadIdx.x;
    if (i < n4) {
        v4f x = ((const v4f*)spre)[i];
        v4f m = ((const v4f*)mu)[i];
        v4f e = ((const v4f*)eps)[i];
        v4f z;
        #pragma unroll
        for (int c = 0; c < 4; ++c) {
            float sp = (x[c] > 20.0f) ? x[c] : log1pf(expf(x[c]));
            z[c] = m[c] + sp * e[c];
        }
        ((v4f*)Z)[i] = z;
    }
}

// ---------------------------------------------------------------------------
// out[row] = sum_j A[row][j]^2   (float4 per lane)
// ---------------------------------------------------------------------------
__global__ __launch_bounds__(256) void rownorm_kernel(
    const float* __restrict__ A, int cols, float* __restrict__ out)
{
    __shared__ float red[256];
    int row = blockIdx.x, tid = threadIdx.x;
    const v4f* A4 = (const v4f*)(A + (size_t)row * cols);
    int cols4 = cols >> 2;
    float s = 0.0f;
    for (int j = tid; j < cols4; j += 256) {
        v4f v = A4[j];
        s += v.x * v.x + v.y * v.y + v.z * v.z + v.w * v.w;
    }
    red[tid] = s;
    __syncthreads();
    for (int off = 128; off; off >>= 1) {
        if (tid < off) red[tid] += red[tid + off];
        __syncthreads();
    }
    if (tid == 0) out[row] = red[0];
}

// ---------------------------------------------------------------------------
// Fine routing: per-row (S=4096) gumbel-softmax + argmax + one-hot + gather
// All global streams are float4 per lane.
// ---------------------------------------------------------------------------
__global__ __launch_bounds__(256) void fine_route_kernel(
    const float* __restrict__ logits, const float* __restrict__ u_s,
    const float* __restrict__ mu_s, const float* __restrict__ sn,
    float* __restrict__ soft_prob, float* __restrict__ hard_prob,
    int* __restrict__ fine_idx, float* __restrict__ qn, float* __restrict__ zq)
{
    __shared__ __align__(16) float sl[S_];
    __shared__ float red[256];
    __shared__ int   redi[256];

    const int row = blockIdx.x, tid = threadIdx.x;
    constexpr int S4 = S_ / 4;
    const v4f* lrow4 = (const v4f*)(logits + (size_t)row * S_);

    for (int j = tid; j < S4; j += 256) *(v4f*)&sl[4 * j] = lrow4[j];
    __syncthreads();

    // pass 1: row max
    float m = -INFINITY;
    for (int j = tid; j < S4; j += 256) {
        v4f v = *(const v4f*)&sl[4 * j];
        m = fmaxf(m, fmaxf(fmaxf(v.x, v.y), fmaxf(v.z, v.w)));
    }
    red[tid] = m; __syncthreads();
    for (int off = 128; off; off >>= 1) {
        if (tid < off) red[tid] = fmaxf(red[tid], red[tid + off]);
        __syncthreads();
    }
    float m1 = red[0]; __syncthreads();

    // pass 2: softmax denom
    float s = 0.0f;
    for (int j = tid; j < S4; j += 256) {
        v4f v = *(const v4f*)&sl[4 * j];
        s += expf(v.x - m1) + expf(v.y - m1) + expf(v.z - m1) + expf(v.w - m1);
    }
    red[tid] = s; __syncthreads();
    for (int off = 128; off; off >>= 1) {
        if (tid < off) red[tid] += red[tid + off];
        __syncthreads();
    }
    float inv_den = 1.0f / red[0]; __syncthreads();

    // pass 3: a = (log(p+1e-9) + gumbel) / 0.5 ; track argmax (first max wins)
    const v4f* urow4 = (const v4f*)(u_s + (size_t)row * S_);
    float am = -INFINITY; int ai = 0x7fffffff;
    for (int j = tid; j < S4; j += 256) {
        v4f v = *(const v4f*)&sl[4 * j];
        v4f u = urow4[j];
        v4f a;
        #pragma unroll
        for (int e = 0; e < 4; ++e) {
            float pp = expf(v[e] - m1) * inv_den;
            float g  = -logf(-logf(u[e] + 1e-20f) + 1e-20f);
            a[e] = (logf(pp + 1e-9f) + g) * 2.0f;   // / temp(0.5)
            if (a[e] > am) { am = a[e]; ai = 4 * j + e; }
        }
        *(v4f*)&sl[4 * j] = a;
    }
    red[tid] = am; redi[tid] = ai; __syncthreads();
    for (int off = 128; off; off >>= 1) {
        if (tid < off) {
            if (red[tid + off] > red[tid] ||
                (red[tid + off] == red[tid] && redi[tid + off] < redi[tid])) {
                red[tid]  = red[tid + off];
                redi[tid] = redi[tid + off];
            }
        }
        __syncthreads();
    }
    float m2 = red[0]; int idx = redi[0]; __syncthreads();

    // pass 4: second softmax denom
    float s2 = 0.0f;
    for (int j = tid; j < S4; j += 256) {
        v4f v = *(const v4f*)&sl[4 * j];
        s2 += expf(v.x - m2) + expf(v.y - m2) + expf(v.z - m2) + expf(v.w - m2);
    }
    red[tid] = s2; __syncthreads();
    for (int off = 128; off; off >>= 1) {
        if (tid < off) red[tid] += red[tid + off];
        __syncthreads();
    }
    float inv2 = 1.0f / red[0]; __syncthreads();

    // outputs are never re-read: non-temporal b128 stores keep L2 for logits
    v4f* sp4 = (v4f*)(soft_prob + (size_t)row * S_);
    v4f* hp4 = (v4f*)(hard_prob + (size_t)row * S_);
    for (int j = tid; j < S4; j += 256) {
        v4f v = *(const v4f*)&sl[4 * j];
        v4f sp, hp;
        #pragma unroll
        for (int e = 0; e < 4; ++e) {
            sp[e] = expf(v[e] - m2) * inv2;
            hp[e] = (4 * j + e == idx) ? 1.0f : 0.0f;
        }
        __builtin_nontemporal_store(sp, &sp4[j]);
        __builtin_nontemporal_store(hp, &hp4[j]);
    }
    // gather selected fine prototype
    const v4f* src4 = (const v4f*)(mu_s + (size_t)idx * D_);
    v4f* dst4 = (v4f*)(zq + (size_t)row * D_);
    for (int k = tid; k < D_ / 4; k += 256) dst4[k] = src4[k];
    if (tid == 0) { fine_idx[row] = idx; qn[row] = sn[idx]; }
}

// ---------------------------------------------------------------------------
// Coarse routing: per-row (Y=256), one element per thread
// ---------------------------------------------------------------------------
__global__ __launch_bounds__(256) void coarse_route_kernel(
    const float* __restrict__ dZ, const float* __restrict__ dQ,
    const float* __restrict__ yn, const float* __restrict__ zn,
    const float* __restrict__ qn, const float* __restrict__ u_y,
    float* __restrict__ soft_y, float* __restrict__ batch_y)
{
    __shared__ float red[256];
    __shared__ int   redi[256];
    const int row = blockIdx.x, j = threadIdx.x;

    // logits = 0.5 * ((2 dZ - zn - yn) + (2 dQ - qn - yn))
    float l = dZ[(size_t)row * Y_ + j] + dQ[(size_t)row * Y_ + j]
              - yn[j] - 0.5f * (zn[row] + qn[row]);

    red[j] = l; __syncthreads();
    for (int off = 128; off; off >>= 1) {
        if (j < off) red[j] = fmaxf(red[j], red[j + off]);
        __syncthreads();
    }
    float m1 = red[0]; __syncthreads();

    float e = expf(l - m1);
    red[j] = e; __syncthreads();
    for (int off = 128; off; off >>= 1) {
        if (j < off) red[j] += red[j + off];
        __syncthreads();
    }
    float p = e / red[0]; __syncthreads();

    float u = u_y[(size_t)row * Y_ + j];
    float g = -logf(-logf(u + 1e-20f) + 1e-20f);
    float a = (logf(p + 1e-9f) + g) * 2.0f;

    red[j] = a; redi[j] = j; __syncthreads();
    for (int off = 128; off; off >>= 1) {
        if (j < off) {
            if (red[j + off] > red[j] ||
                (red[j + off] == red[j] && redi[j + off] < redi[j])) {
                red[j]  = red[j + off];
                redi[j] = redi[j + off];
            }
        }
        __syncthreads();
    }
    float m2 = red[0]; int idx = redi[0]; __syncthreads();

    float e2 = expf(a - m2);
    red[j] = e2; __syncthreads();
    for (int off = 128; off; off >>= 1) {
        if (j < off) red[j] += red[j + off];
        __syncthreads();
    }
    __builtin_nontemporal_store(e2 / red[0], &soft_y[(size_t)row * Y_ + j]);
    if (j == 0) batch_y[row] = (float)idx;
}

// ---------------------------------------------------------------------------
extern "C" void kernel_launch(void* const* d_in, const int* in_sizes, int n_in,
                              void* d_out, int out_size, void* d_ws, size_t ws_size,
                              hipStream_t stream)
{
    const float* X    = (const float*)d_in[0];
    const float* W1   = (const float*)d_in[1];
    const float* b1   = (const float*)d_in[2];
    const float* W2   = (const float*)d_in[3];
    const float* b2   = (const float*)d_in[4];
    const float* Ws1  = (const float*)d_in[5];
    const float* bs1  = (const float*)d_in[6];
    const float* Ws2  = (const float*)d_in[7];
    const float* bs2  = (const float*)d_in[8];
    const float* mu_y = (const float*)d_in[9];
    const float* mu_s = (const float*)d_in[10];
    const float* eps  = (const float*)d_in[11];
    const float* u_s  = (const float*)d_in[12];
    const float* u_y  = (const float*)d_in[13];

    // output layout (flat, return order)
    float* out   = (float*)d_out;
    float* o_mu  = out;                                   // [B,D]
    float* o_Z   = o_mu + (size_t)B_ * D_;                // [B,D]
    float* o_sp  = o_Z + (size_t)B_ * D_;                 // [B,S]
    float* o_spy = o_sp + (size_t)B_ * S_;                // [B,Y]
    float* o_hp  = o_spy + (size_t)B_ * Y_;               // [B,S]
    float* o_by  = o_hp + (size_t)B_ * S_;                // [B]

    // workspace layout (floats). H1/Hs1/spre are dead before logits is written,
    // so the logits buffer aliases them.
    float* ws     = (float*)d_ws;
    float* H1     = ws;                                   // [B,H]   32 MB
    float* Hs1    = ws + (size_t)B_ * H_;                 // [B,H]   32 MB
    float* spre   = ws + 2 * (size_t)B_ * H_;             // [B,D]   16 MB
    float* logits = ws;                                   // [B,S]  128 MB (alias)
    size_t p = (size_t)B_ * S_;
    float* zn = ws + p; p += B_;
    float* sn = ws + p; p += S_;
    float* yn = ws + p; p += Y_;
    float* qn = ws + p; p += B_;
    float* zq = ws + p; p += (size_t)B_ * D_;
    float* dZ = ws + p; p += (size_t)B_ * Y_;
    float* dQ = ws + p; p += (size_t)B_ * Y_;
    int*  fidx = (int*)(ws + p); p += B_;

    const dim3 blk(256);

    // 1) first MLP layers (relu epilogue)
    gemm_wmma_f32<1, false><<<dim3(B_ / BM, H_ / BN), blk, 0, stream>>>(
        X, W1, b1, H1, B_, H_, D_, nullptr, nullptr);
    gemm_wmma_f32<1, false><<<dim3(B_ / BM, H_ / BN), blk, 0, stream>>>(
        X, Ws1, bs1, Hs1, B_, H_, D_, nullptr, nullptr);

    // 2) second MLP layers
    gemm_wmma_f32<0, false><<<dim3(B_ / BM, D_ / BN), blk, 0, stream>>>(
        H1, W2, b2, o_mu, B_, D_, H_, nullptr, nullptr);
    gemm_wmma_f32<0, false><<<dim3(B_ / BM, D_ / BN), blk, 0, stream>>>(
        Hs1, Ws2, bs2, spre, B_, D_, H_, nullptr, nullptr);

    // 3) reparameterization (float4 per lane)
    z_kernel<<<(B_ * D_ / 4) / 256, 256, 0, stream>>>(o_mu, spre, eps, o_Z,
                                                      B_ * D_ / 4);

    // 4) squared norms
    rownorm_kernel<<<B_, 256, 0, stream>>>(o_Z, D_, zn);
    rownorm_kernel<<<S_, 256, 0, stream>>>(mu_s, D_, sn);
    rownorm_kernel<<<Y_, 256, 0, stream>>>(mu_y, D_, yn);

    // 5) fine logits: 2*Z@mu_s^T - ||Z||^2 - ||mu_s||^2
    gemm_wmma_f32<2, true><<<dim3(B_ / BM, S_ / BN), blk, 0, stream>>>(
        o_Z, mu_s, nullptr, logits, B_, S_, D_, zn, sn);

    // 6) fine routing (soft_prob, hard_prob, fine_idx, z_q, qn)
    fine_route_kernel<<<B_, 256, 0, stream>>>(
        logits, u_s, mu_s, sn, o_sp, o_hp, fidx, qn, zq);

    // 7) coarse dot products
    gemm_wmma_f32<3, true><<<dim3(B_ / BM, Y_ / BN), blk, 0, stream>>>(
        o_Z, mu_y, nullptr, dZ, B_, Y_, D_, nullptr, nullptr);
    gemm_wmma_f32<3, true><<<dim3(B_ / BM, Y_ / BN), blk, 0, stream>>>(
        zq, mu_y, nullptr, dQ, B_, Y_, D_, nullptr, nullptr);

    // 8) coarse routing (soft_prob_y, batch_y)
    coarse_route_kernel<<<B_, 256, 0, stream>>>(
        dZ, dQ, yn, zn, qn, u_y, o_spy, o_by);
}